// CenterEstimator_14551349199509
// MI455X (gfx1250) — compile-verified
//
#include <hip/hip_runtime.h>
#include <hip/hip_bf16.h>

// ---------------------------------------------------------------------------
// CenterEstimator for MI455X (gfx1250, wave32).
// Bandwidth-bound (~230 MB HBM traffic -> ~10-11us floor @ 23.3 TB/s).
// Pass 1: fused copy/exp + 6 dilated 3x3 stencils. Interior tiles stage the
//         input planes with GLOBAL_LOAD_ASYNC_TO_LDS_B128 (ASYNCcnt path,
//         no VGPR bounce); edge tiles use a zero-padded scalar path.
// Pass 2: 3x3 box filter as banded GEMM on V_WMMA_F32_16X16X4_F32 chains,
//         max-pool peak test, atomic compaction of candidates.
// Pass 3: per-batch radix-select + bitonic sort -> top-2048 center rows.
// ---------------------------------------------------------------------------

typedef __attribute__((ext_vector_type(2))) float v2f;
typedef __attribute__((ext_vector_type(8))) float v8f;

#define HW 1024
#define NB 8
#define PLANE (HW * HW)
#define MAXC 2048
#define CAP (1 << 18)            // per-batch candidate capacity (16 MB total in ws)
#define NEG_INF_F (-1000000000.0f)
#define THR 0.1f
#define BORDER_LO 5
#define BORDER_HI (HW - 1 - BORDER_LO)   // inclusive

// LDS byte offset of a __shared__ object: flat shared addr = {aperture, off32}.
__device__ __forceinline__ unsigned lds_offset(const void* p) {
    return (unsigned)(size_t)p;
}

// ============================================================================
// Kernel 1: output planes (S, C, exp(R)-1) + conv_resp (relu of 6 dilated convs)
// Block: 256 threads, 32x32 output tile, halo 16 -> 64x64 LDS region per plane.
// Row stride 68 floats keeps every B128 async LDS write 16-byte aligned.
// ============================================================================
__global__ __launch_bounds__(256)
void front_kernel(const float* __restrict__ in,
                  const float* __restrict__ w_c,
                  const float* __restrict__ w_s,
                  float* __restrict__ out0,
                  float* __restrict__ convr) {
    __shared__ __align__(16) float sA[64][68];   // channel 0 (S)
    __shared__ __align__(16) float sB[64][68];   // channel 1 (C)

    const int b  = blockIdx.z;
    const int R0 = blockIdx.y * 32;
    const int C0 = blockIdx.x * 32;
    const int tid = threadIdx.x;

    const float* pS = in + ((size_t)b * 3 + 0) * PLANE;
    const float* pC = in + ((size_t)b * 3 + 1) * PLANE;
    const float* pR = in + ((size_t)b * 3 + 2) * PLANE;

    const bool interior = (R0 >= 16) && (R0 + 48 <= HW) &&
                          (C0 >= 16) && (C0 + 48 <= HW);

    if (interior) {
        // Async DMA path: 64x64 region per plane as 16-byte chunks,
        // memory -> LDS directly (no VGPR staging), tracked on ASYNCcnt.
        for (int l = tid; l < 64 * 16; l += 256) {
            int i  = l >> 4;             // region row 0..63
            int j4 = (l & 15) << 2;      // starting float col 0..60
            size_t off = (size_t)(R0 - 16 + i) * HW + (C0 - 16 + j4);
            unsigned la = lds_offset(&sA[i][j4]);
            unsigned lb = lds_offset(&sB[i][j4]);
            unsigned long long ga = (unsigned long long)(pS + off);
            unsigned long long gb = (unsigned long long)(pC + off);
            asm volatile("global_load_async_to_lds_b128 %0, %1, off"
                         :: "v"(la), "v"(ga) : "memory");
            asm volatile("global_load_async_to_lds_b128 %0, %1, off"
                         :: "v"(lb), "v"(gb) : "memory");
        }
        asm volatile("s_wait_asynccnt 0x0" ::: "memory");
    } else {
        // Edge tiles: zero-padded ('SAME' conv) scalar path.
        for (int l = tid; l < 64 * 64; l += 256) {
            int i = l >> 6, j = l & 63;
            int gy = R0 - 16 + i, gx = C0 - 16 + j;
            bool ok = (gy >= 0) && (gy < HW) && (gx >= 0) && (gx < HW);
            size_t off = (size_t)gy * HW + gx;
            sA[i][j] = ok ? pS[off] : 0.0f;
            sB[i][j] = ok ? pC[off] : 0.0f;
        }
    }
    __syncthreads();

    float* oS = out0 + ((size_t)b * 3 + 0) * PLANE;
    float* oC = out0 + ((size_t)b * 3 + 1) * PLANE;
    float* oR = out0 + ((size_t)b * 3 + 2) * PLANE;
    float* cr = convr + (size_t)b * PLANE;

    const int DIL[3] = {1, 4, 16};

    for (int k = 0; k < 4; ++k) {
        int p  = tid + 256 * k;
        int ly = p >> 5, lx = p & 31;
        int y = 16 + ly, x = 16 + lx;
        int gy = R0 + ly, gx = C0 + lx;
        size_t off = (size_t)gy * HW + gx;

        float acc = 0.0f;
#pragma unroll
        for (int di = 0; di < 3; ++di) {
            int d = DIL[di];
#pragma unroll
            for (int j = 0; j < 3; ++j)
#pragma unroll
                for (int i = 0; i < 3; ++i) {
                    // w_c convolves C (plane 1), w_s convolves S (plane 0)
                    acc += w_c[di * 9 + j * 3 + i] * sB[y + (j - 1) * d][x + (i - 1) * d];
                    acc += w_s[di * 9 + j * 3 + i] * sA[y + (j - 1) * d][x + (i - 1) * d];
                }
        }
        float resp = fmaxf(acc, 0.0f);

        oS[off] = sA[y][x];
        oC[off] = sB[y][x];
        oR[off] = expf(pR[off]) - 1.0f;
        cr[off] = resp;
    }
}

// ============================================================================
// Kernel 2: smooth = avg3(conv_resp) via banded WMMA GEMMs, peak test,
//           candidate compaction. Block: 128 threads (4 waves).
// WG computes smooth over a 32x32 region, emits scores for the inner 30x30.
// smooth local (r,c) <-> global (R0-1+r, C0-1+c).
// X (conv_resp) region: rows/cols R0-2..R0+31 -> local 0..33.
// Vertical:   T[r][j] = sum_{k=r..r+2} X[k][j]       (A = banded ones, K=36)
// Horizontal: S[r][c] = sum_{k=c..c+2} T[r][k] / 9   (B = banded ones, K=36)
// ============================================================================
__global__ __launch_bounds__(128)
void peaks_kernel(const float* __restrict__ convr,
                  unsigned* __restrict__ counters,
                  uint2* __restrict__ cands) {
    __shared__ float Xs[36][48];   // conv_resp region (zero padded)
    __shared__ float Ts[32][48];   // vertical pass result
    __shared__ float Ss[32][36];   // smooth 32x32

    const int b  = blockIdx.z;
    const int R0 = blockIdx.y * 30;
    const int C0 = blockIdx.x * 30;
    const int tid = threadIdx.x;
    const float* cr = convr + (size_t)b * PLANE;

    // Zero-fill Xs (covers K-padding rows/cols), then load the 34x34 region.
    for (int l = tid; l < 36 * 48; l += 128) (&Xs[0][0])[l] = 0.0f;
    __syncthreads();
    for (int l = tid; l < 34 * 34; l += 128) {
        int i = l / 34, j = l % 34;
        int gy = R0 - 2 + i, gx = C0 - 2 + j;
        bool ok = (gy >= 0) && (gy < HW) && (gx >= 0) && (gx < HW);
        Xs[i][j] = ok ? cr[(size_t)gy * HW + gx] : 0.0f;
    }
    __syncthreads();

    const int wv   = tid >> 5;        // wave id 0..3
    const int lane = tid & 31;
    const int ln   = lane & 15;       // row/col index within fragment
    const int half = lane >> 4;       // selects K-pair {0,1} vs {2,3}

    // ---- Vertical banded GEMM: 2 (M) x 3 (N) tiles over 4 waves -------------
    for (int job = wv; job < 6; job += 4) {
        int tm = job / 3, tn = job % 3;
        int Rm = tm * 16 + ln;        // output row this lane holds in A
        int Cn = tn * 16 + ln;        // B / D column
        v8f acc = {};
#pragma unroll
        for (int kk = 0; kk < 9; ++kk) {
            int k0 = kk * 4 + half * 2;
            v2f a, bf;
            a.x  = (k0     >= Rm && k0     <= Rm + 2) ? 1.0f : 0.0f;
            a.y  = (k0 + 1 >= Rm && k0 + 1 <= Rm + 2) ? 1.0f : 0.0f;
            bf.x = Xs[k0][Cn];
            bf.y = Xs[k0 + 1][Cn];
            acc = __builtin_amdgcn_wmma_f32_16x16x4_f32(
                false, a, false, bf, (short)0, acc, false, false);
        }
#pragma unroll
        for (int g = 0; g < 8; ++g)
            Ts[tm * 16 + g + half * 8][tn * 16 + ln] = acc[g];
    }
    __syncthreads();

    // ---- Horizontal banded GEMM: 2x2 tiles, one per wave --------------------
    {
        int sm = wv >> 1, sn = wv & 1;
        int Rm = sm * 16 + ln;
        int Cn = sn * 16 + ln;
        v8f acc = {};
#pragma unroll
        for (int kk = 0; kk < 9; ++kk) {
            int k0 = kk * 4 + half * 2;
            v2f a, bf;
            a.x  = Ts[Rm][k0];
            a.y  = Ts[Rm][k0 + 1];
            bf.x = (k0     >= Cn && k0     <= Cn + 2) ? 1.0f : 0.0f;
            bf.y = (k0 + 1 >= Cn && k0 + 1 <= Cn + 2) ? 1.0f : 0.0f;
            acc = __builtin_amdgcn_wmma_f32_16x16x4_f32(
                false, a, false, bf, (short)0, acc, false, false);
        }
        const float inv9 = 1.0f / 9.0f;
#pragma unroll
        for (int g = 0; g < 8; ++g)
            Ss[sm * 16 + g + half * 8][sn * 16 + ln] = acc[g] * inv9;
    }
    __syncthreads();

    // ---- Peak test on inner 30x30 core --------------------------------------
    for (int p = tid; p < 30 * 30; p += 128) {
        int r = 1 + p / 30, c = 1 + p % 30;     // local smooth coords
        int gy = R0 - 1 + r, gx = C0 - 1 + c;   // == R0..R0+29
        if (gy >= HW || gx >= HW) continue;

        float sm = Ss[r][c];
        float mx = sm;
#pragma unroll
        for (int dy = -1; dy <= 1; ++dy)
#pragma unroll
            for (int dx = -1; dx <= 1; ++dx)
                mx = fmaxf(mx, Ss[r + dy][c + dx]);

        bool peak = (sm >= mx) && (sm > THR) &&
                    (gy >= BORDER_LO) && (gy <= BORDER_HI) &&
                    (gx >= BORDER_LO) && (gx <= BORDER_HI);
        float cv = Xs[r + 1][c + 1];             // conv_resp at (gy,gx)
        if (peak && cv > THR) {
            unsigned pos = atomicAdd(&counters[b], 1u);
            if (pos < CAP)
                cands[(size_t)b * CAP + pos] =
                    make_uint2(__float_as_uint(cv), (unsigned)(gy * HW + gx));
        }
    }
}

// ============================================================================
// Kernel 3: per-batch top-2048 (radix-select on float bits + bitonic sort).
// One 1024-thread block per batch.
// ============================================================================
__device__ __forceinline__ bool before_valdesc(uint2 a, uint2 b) {
    float av = __uint_as_float(a.x), bv = __uint_as_float(b.x);
    if (av > bv) return true;
    if (av < bv) return false;
    return a.y < b.y;                 // ties: lower index first (lax.top_k)
}
__device__ __forceinline__ bool before_idx(uint2 a, uint2 b) { return a.y < b.y; }

__device__ void bitonic_sort_2048(uint2* arr, int tid, int mode) {
    for (int k = 2; k <= 2048; k <<= 1) {
        for (int j = k >> 1; j > 0; j >>= 1) {
            __syncthreads();
            for (int t = tid; t < 2048; t += 1024) {
                int pr = t ^ j;
                if (pr > t) {
                    bool up = ((t & k) == 0);
                    uint2 x = arr[t], y = arr[pr];
                    bool bf = mode ? before_idx(x, y) : before_valdesc(x, y);
                    if (up ? !bf : bf) { arr[t] = y; arr[pr] = x; }
                }
            }
        }
    }
    __syncthreads();
}

__device__ unsigned block_count_ge(const uint2* cb, unsigned N, unsigned u,
                                   unsigned* red, int tid) {
    unsigned c = 0;
    for (unsigned i = tid; i < N; i += 1024) c += (cb[i].x >= u) ? 1u : 0u;
    red[tid] = c;
    __syncthreads();
    for (int s = 512; s > 0; s >>= 1) {
        if (tid < s) red[tid] += red[tid + s];
        __syncthreads();
    }
    unsigned tot = red[0];
    __syncthreads();
    return tot;
}

__global__ __launch_bounds__(1024)
void topk_kernel(const uint2* __restrict__ cands,
                 const unsigned* __restrict__ counters,
                 float* __restrict__ cp) {
    __shared__ uint2 list[2048];
    __shared__ uint2 ties[2048];
    __shared__ unsigned red[1024];
    __shared__ unsigned sh_nsel, sh_ntie;

    const int b = blockIdx.x;
    const int tid = threadIdx.x;
    const uint2* cb = cands + (size_t)b * CAP;
    unsigned N = counters[b];
    if (N > CAP) N = CAP;

    unsigned total;
    if (N > (unsigned)MAXC) {
        // Radix-select kth-largest value bits (all candidates positive floats).
        unsigned u = 0;
        for (int bit = 30; bit >= 0; --bit) {
            unsigned cu = u | (1u << bit);
            if (block_count_ge(cb, N, cu, red, tid) >= (unsigned)MAXC) u = cu;
        }
        unsigned G = block_count_ge(cb, N, u + 1, red, tid);   // strictly greater

        if (tid == 0) { sh_nsel = 0; sh_ntie = 0; }
        __syncthreads();
        for (unsigned i = tid; i < N; i += 1024) {
            uint2 e = cb[i];
            if (e.x > u) {
                unsigned p = atomicAdd(&sh_nsel, 1u);
                list[p] = e;
            } else if (e.x == u) {
                unsigned p = atomicAdd(&sh_ntie, 1u);
                if (p < 2048) ties[p] = e;
            }
        }
        __syncthreads();
        unsigned E = sh_ntie; if (E > 2048) E = 2048;
        for (int t = tid; t < 2048; t += 1024)
            if ((unsigned)t >= E) ties[t] = make_uint2(0u, 0xFFFFFFFFu);
        bitonic_sort_2048(ties, tid, 1);                       // by idx asc
        unsigned need = (unsigned)MAXC - G;
        for (unsigned t = tid; t < need; t += 1024) list[G + t] = ties[t];
        total = MAXC;
    } else {
        for (unsigned i = tid; i < N; i += 1024) list[i] = cb[i];
        total = N;
    }
    __syncthreads();
    for (int t = tid; t < 2048; t += 1024)
        if ((unsigned)t >= total)
            list[t] = make_uint2(__float_as_uint(-1.0e30f), 0x7FFFFFFFu);
    bitonic_sort_2048(list, tid, 0);                           // val desc, idx asc

    for (int t = tid; t < 2048; t += 1024) {
        uint2 e = list[t];
        float v = __uint_as_float(e.x);
        bool valid = ((unsigned)t < total) && (v > THR);
        float* row = cp + ((size_t)b * MAXC + t) * 5;
        row[0] = valid ? 1.0f : 0.0f;
        row[1] = valid ? (float)(e.y % (unsigned)HW) : 0.0f;
        row[2] = valid ? (float)(e.y / (unsigned)HW) : 0.0f;
        row[3] = valid ? v : 0.0f;
        row[4] = valid ? v : 0.0f;
    }
}

__global__ void init_kernel(unsigned* counters) {
    if (threadIdx.x < NB) counters[threadIdx.x] = 0u;
}

// ============================================================================
extern "C" void kernel_launch(void* const* d_in, const int* in_sizes, int n_in,
                              void* d_out, int out_size, void* d_ws, size_t ws_size,
                              hipStream_t stream) {
    const float* in  = (const float*)d_in[0];
    const float* w_c = (const float*)d_in[1];   // (3,1,1,3,3) = 27 floats
    const float* w_s = (const float*)d_in[2];

    float* out0  = (float*)d_out;                         // (8,3,1024,1024)
    float* cp    = out0 + (size_t)NB * 3 * PLANE;         // (8,2048,5)
    float* convr = cp + (size_t)NB * MAXC * 5;            // (8,1,1024,1024)

    unsigned* counters = (unsigned*)d_ws;                 // 8 counters
    uint2* cands = (uint2*)((char*)d_ws + 256);           // 8 * CAP * uint2

    init_kernel<<<1, 32, 0, stream>>>(counters);
    front_kernel<<<dim3(32, 32, NB), 256, 0, stream>>>(in, w_c, w_s, out0, convr);
    peaks_kernel<<<dim3(35, 35, NB), 128, 0, stream>>>(convr, counters, cands);
    topk_kernel<<<NB, 1024, 0, stream>>>(cands, counters, cp);
}